// LSTMLayerCustom_77017353552602
// MI455X (gfx1250) — compile-verified
//
#include <hip/hip_runtime.h>

typedef __attribute__((ext_vector_type(2))) float v2f;
typedef __attribute__((ext_vector_type(4))) float v4f;
typedef __attribute__((ext_vector_type(8))) float v8f;

#define BDIM 64
#define TDIM 512
#define IDIM 512
#define HDIM 512
#define NGATE 2048   // 4*H

__device__ __forceinline__ float sigm(float x) { return 1.0f / (1.0f + __expf(-x)); }

// ---------------------------------------------------------------------------
// Phase 1: x_proj[t][b][n] = sum_k X[b][t][k] * W_ih[n][k] + b_ih[n] + b_hh[n]
//   (stored TIME-MAJOR so each recurrent step reads one contiguous 512KB slab)
//   M = B*T = 32768 (m = b*T + t), N = 4H = 2048, K = I = 512
// Block (256 thr = 8 waves) computes a 128(M) x 64(N) tile; W_ih panel staged
// in LDS in K-chunks of 64 (stride 68 -> conflict-free ds_load_b64 fragments).
// Grid: 256 mBlocks * 32 nStrips = 8192 blocks.
// ---------------------------------------------------------------------------
#define XP_CHUNK 64
#define XP_STRIDE (XP_CHUNK + 4)

__global__ void xproj_kernel(const float* __restrict__ X,
                             const float* __restrict__ Wih,
                             const float* __restrict__ bih,
                             const float* __restrict__ bhh,
                             float* __restrict__ xp) {
    __shared__ float ldsB[64 * XP_STRIDE];

    const int K = IDIM;
    const int lane = threadIdx.x & 31;
    const int wave = threadIdx.x >> 5;           // 0..7

    const int NSTRIPS = NGATE / 64;              // 32
    const int mBlock = blockIdx.x / NSTRIPS;     // 0..255
    const int nStrip = blockIdx.x % NSTRIPS;     // 0..31
    const int nBase  = nStrip * 64;
    const int mBase  = mBlock * 128;

    const int nCol  = lane & 15;
    const int khalf = (lane >> 4) * 2;           // 0 or 2
    const int mRow  = mBase + wave * 16 + (lane & 15);

    v8f acc[4];
    const v8f zero = {0.f,0.f,0.f,0.f,0.f,0.f,0.f,0.f};
#pragma unroll
    for (int j = 0; j < 4; ++j) acc[j] = zero;

    const float* __restrict__ arow = X + (size_t)mRow * K;

    const float* bcol[4];
#pragma unroll
    for (int j = 0; j < 4; ++j) bcol[j] = &ldsB[(j * 16 + nCol) * XP_STRIDE];

    for (int kc = 0; kc < K; kc += XP_CHUNK) {
        __syncthreads();
#pragma unroll
        for (int it = 0; it < 4; ++it) {
            const int i   = threadIdx.x + it * 256;
            const int col = i >> 4;               // 0..63
            const int f4  = i & 15;               // 0..15
            const v4f d = *(const v4f*)(Wih + (size_t)(nBase + col) * K + kc + f4 * 4);
            *(v4f*)&ldsB[col * XP_STRIDE + f4 * 4] = d;
        }
        __syncthreads();

#pragma unroll 4
        for (int k = 0; k < XP_CHUNK; k += 4) {
            const int kb = k + khalf;
            v2f a = *(const v2f*)(arow + kc + kb);
#pragma unroll
            for (int j = 0; j < 4; ++j) {
                v2f b = *(const v2f*)(bcol[j] + kb);
                acc[j] = __builtin_amdgcn_wmma_f32_16x16x4_f32(
                    false, a, false, b, (short)0, acc[j], false, false);
            }
        }
    }

    const int rowHalf = (lane >> 4) * 8;
#pragma unroll
    for (int j = 0; j < 4; ++j) {
        const int n = nBase + j * 16 + nCol;
        const float bias = bih[n] + bhh[n];
#pragma unroll
        for (int v = 0; v < 8; ++v) {
            const int mm = mBase + wave * 16 + rowHalf + v;  // mm = b*T + t
            const int b  = mm >> 9;                          // /T
            const int tt = mm & (TDIM - 1);                  // %T
            xp[((size_t)tt * BDIM + b) * NGATE + n] = acc[j][v] + bias;
        }
    }
}

// ---------------------------------------------------------------------------
// Barrier counter init (stream-ordered, graph-safe).
// ---------------------------------------------------------------------------
__global__ void init_barrier_kernel(unsigned* bar) {
    if (threadIdx.x == 0) *bar = 0u;
}

// ---------------------------------------------------------------------------
// Phase 2: PERSISTENT recurrence. One block per jTile (16 hidden cols), 32
// blocks total (one WGP each -> co-resident, deadlock-free). Each block
// stages its 64 W_hh columns (4 gates x 16 cols, K=512; ~129KB of the WGP's
// 320KB LDS) ONCE, then iterates all T=512 steps internally:
//   gates = xp[t] + h_{t-1} @ Whh_panel   (4 fp32-WMMA accumulator chains)
// c for the block's own columns lives in registers across the whole sequence;
// h is exchanged through global memory with a monotonic-counter grid barrier.
// ---------------------------------------------------------------------------
#define PS_STRIDE (HDIM + 4)                    // 516 floats per LDS column
#define PS_LDS_BYTES (64 * PS_STRIDE * 4)       // 132096 B

__global__ void lstm_persistent_kernel(const float* __restrict__ xp,
                                       const float* __restrict__ Whh,
                                       float* __restrict__ hs,
                                       float* __restrict__ cs,
                                       unsigned* __restrict__ bar) {
    extern __shared__ float ldsB[];

    const int lane  = threadIdx.x & 31;
    const int wave  = threadIdx.x >> 5;          // 0..3 -> mTile (16 batches)
    const int jTile = blockIdx.x;                // 0..31
    const int jBase = jTile * 16;

    const int nCol  = lane & 15;
    const int khalf = (lane >> 4) * 2;
    const int mTile = wave;
    const int mRowA = mTile * 16 + (lane & 15);  // batch row for A fragment

    // ---- stage the full W_hh panel once: 64 cols x 128 float4 ----
    for (int i = threadIdx.x; i < 64 * 128; i += 128) {
        const int col = i >> 7;                  // 0..63
        const int f4  = i & 127;                 // 0..127
        const int g   = col >> 4;                // gate
        const int c16 = col & 15;
        const int n   = g * HDIM + jBase + c16;
        const v4f d = *(const v4f*)(Whh + (size_t)n * HDIM + f4 * 4);
        *(v4f*)&ldsB[col * PS_STRIDE + f4 * 4] = d;
    }
    __syncthreads();

    const float* bcol[4];
#pragma unroll
    for (int g = 0; g < 4; ++g) bcol[g] = &ldsB[(g * 16 + nCol) * PS_STRIDE + khalf];

    const int rowHalf = (lane >> 4) * 8;
    const int j = jBase + nCol;
    const unsigned nBlocks = gridDim.x;
    const v8f zero = {0.f,0.f,0.f,0.f,0.f,0.f,0.f,0.f};

    float creg[8];
#pragma unroll
    for (int v = 0; v < 8; ++v) creg[v] = 0.0f;

    for (int t = 0; t < TDIM; ++t) {
        // issue xp loads early; their waits sink below the 512-WMMA K-loop
        float xg[4][8];
#pragma unroll
        for (int v = 0; v < 8; ++v) {
            const int mm = mTile * 16 + rowHalf + v;
            const float* __restrict__ xpr = xp + ((size_t)t * BDIM + mm) * NGATE;
            xg[0][v] = xpr[0 * HDIM + j];
            xg[1][v] = xpr[1 * HDIM + j];
            xg[2][v] = xpr[2 * HDIM + j];
            xg[3][v] = xpr[3 * HDIM + j];
        }

        v8f acc[4];
#pragma unroll
        for (int g = 0; g < 4; ++g) acc[g] = zero;

        if (t > 0) {
            const float* __restrict__ hrow =
                hs + ((size_t)mRowA * TDIM + (t - 1)) * HDIM;
#pragma unroll 4
            for (int k = 0; k < HDIM; k += 4) {
                v2f a = *(const v2f*)(hrow + k + khalf);
#pragma unroll
                for (int g = 0; g < 4; ++g) {
                    v2f b = *(const v2f*)(bcol[g] + k);
                    acc[g] = __builtin_amdgcn_wmma_f32_16x16x4_f32(
                        false, a, false, b, (short)0, acc[g], false, false);
                }
            }
        }

#pragma unroll
        for (int v = 0; v < 8; ++v) {
            const int mm = mTile * 16 + rowHalf + v;       // batch index
            const float ig = sigm(xg[0][v] + acc[0][v]);
            const float fg = sigm(xg[1][v] + acc[1][v]);
            const float g_ = tanhf(xg[2][v] + acc[2][v]);
            const float og = sigm(xg[3][v] + acc[3][v]);

            const float cnew = fg * creg[v] + ig * g_;
            creg[v] = cnew;
            const float hnew = og * tanhf(cnew);

            hs[((size_t)mm * TDIM + t) * HDIM + j] = hnew;
            cs[((size_t)mm * TDIM + t) * HDIM + j] = cnew;
        }

        // ---- grid barrier: release h_t, wait for all 32 blocks ----
        __threadfence();
        __syncthreads();
        if (threadIdx.x == 0) {
            __hip_atomic_fetch_add(bar, 1u, __ATOMIC_RELEASE, __HIP_MEMORY_SCOPE_AGENT);
            const unsigned target = (unsigned)(t + 1) * nBlocks;
            while (__hip_atomic_load(bar, __ATOMIC_ACQUIRE, __HIP_MEMORY_SCOPE_AGENT) < target) {
                __builtin_amdgcn_s_sleep(1);
            }
        }
        __syncthreads();
        __threadfence();
    }
}

// ---------------------------------------------------------------------------
// Fallback (if d_ws cannot hold x_proj): fully fused per-step kernel,
// gates = [x_t | h_{t-1}] @ [W_ih ; W_hh]^T + b_ih + b_hh
// ---------------------------------------------------------------------------
__global__ void lstm_step_fused_kernel(const float* __restrict__ X,
                                       const float* __restrict__ Wih,
                                       const float* __restrict__ Whh,
                                       const float* __restrict__ bih,
                                       const float* __restrict__ bhh,
                                       float* __restrict__ hs,
                                       float* __restrict__ cs,
                                       int t) {
    const int lane   = threadIdx.x & 31;
    const int waveId = blockIdx.x * (blockDim.x >> 5) + (threadIdx.x >> 5);

    const int mTile = waveId >> 5;
    const int jTile = waveId & 31;
    const int jBase = jTile * 16;
    const int nCol  = lane & 15;
    const int khalf = (lane >> 4) * 2;
    const int mRow  = mTile * 16 + (lane & 15);

    v8f acc[4];
    const v8f zero = {0.f,0.f,0.f,0.f,0.f,0.f,0.f,0.f};
#pragma unroll
    for (int g = 0; g < 4; ++g) acc[g] = zero;

    {
        const float* __restrict__ xrow = X + ((size_t)mRow * TDIM + t) * IDIM;
        for (int k = 0; k < IDIM; k += 4) {
            const int kb = k + khalf;
            v2f a = *(const v2f*)(xrow + kb);
#pragma unroll
            for (int g = 0; g < 4; ++g) {
                const int n = g * HDIM + jBase + nCol;
                v2f b = *(const v2f*)(Wih + (size_t)n * IDIM + kb);
                acc[g] = __builtin_amdgcn_wmma_f32_16x16x4_f32(
                    false, a, false, b, (short)0, acc[g], false, false);
            }
        }
    }
    if (t > 0) {
        const float* __restrict__ hrow = hs + ((size_t)mRow * TDIM + (t - 1)) * HDIM;
        for (int k = 0; k < HDIM; k += 4) {
            const int kb = k + khalf;
            v2f a = *(const v2f*)(hrow + kb);
#pragma unroll
            for (int g = 0; g < 4; ++g) {
                const int n = g * HDIM + jBase + nCol;
                v2f b = *(const v2f*)(Whh + (size_t)n * HDIM + kb);
                acc[g] = __builtin_amdgcn_wmma_f32_16x16x4_f32(
                    false, a, false, b, (short)0, acc[g], false, false);
            }
        }
    }

    const int rowHalf = (lane >> 4) * 8;
    const int j = jBase + nCol;
#pragma unroll
    for (int v = 0; v < 8; ++v) {
        const int mm = mTile * 16 + rowHalf + v;
        const float bi = bih[0 * HDIM + j] + bhh[0 * HDIM + j];
        const float bf = bih[1 * HDIM + j] + bhh[1 * HDIM + j];
        const float bg = bih[2 * HDIM + j] + bhh[2 * HDIM + j];
        const float bo = bih[3 * HDIM + j] + bhh[3 * HDIM + j];

        const float ig = sigm(acc[0][v] + bi);
        const float fg = sigm(acc[1][v] + bf);
        const float g_ = tanhf(acc[2][v] + bg);
        const float og = sigm(acc[3][v] + bo);

        const float cprev = (t > 0) ? cs[((size_t)mm * TDIM + (t - 1)) * HDIM + j] : 0.0f;
        const float cnew  = fg * cprev + ig * g_;
        const float hnew  = og * tanhf(cnew);

        hs[((size_t)mm * TDIM + t) * HDIM + j] = hnew;
        cs[((size_t)mm * TDIM + t) * HDIM + j] = cnew;
    }
}

extern "C" void kernel_launch(void* const* d_in, const int* in_sizes, int n_in,
                              void* d_out, int out_size, void* d_ws, size_t ws_size,
                              hipStream_t stream) {
    const float* x   = (const float*)d_in[0];
    const float* Wih = (const float*)d_in[1];
    const float* Whh = (const float*)d_in[2];
    const float* bih = (const float*)d_in[3];
    const float* bhh = (const float*)d_in[4];

    float* hs = (float*)d_out;
    float* cs = hs + (size_t)BDIM * TDIM * HDIM;

    const size_t xp_bytes = (size_t)BDIM * TDIM * NGATE * sizeof(float);  // 256 MB

    if (ws_size >= xp_bytes + 256) {
        float*    xp  = (float*)d_ws;
        unsigned* bar = (unsigned*)((char*)d_ws + xp_bytes);

        init_barrier_kernel<<<1, 32, 0, stream>>>(bar);
        xproj_kernel<<<8192, 256, 0, stream>>>(x, Wih, bih, bhh, xp);
        lstm_persistent_kernel<<<32, 128, PS_LDS_BYTES, stream>>>(xp, Whh, hs, cs, bar);
    } else {
        for (int t = 0; t < TDIM; ++t) {
            lstm_step_fused_kernel<<<32, 128, 0, stream>>>(x, Wih, Whh, bih, bhh, hs, cs, t);
        }
    }
}